// ImageAlignmentModel_18116172054597
// MI455X (gfx1250) — compile-verified
//
#include <hip/hip_runtime.h>
#include <hip/hip_bf16.h>
#include <stdint.h>

#define TPB 256
#define PTS_PER_THREAD 4
#define PTS_PER_TILE (TPB * PTS_PER_THREAD)          // 1024 points / tile
#define CHUNKS_PER_TILE (PTS_PER_TILE * 3 / 4)       // 768 float4 (12 KB)
#define WAVES_PER_BLOCK (TPB / 32)                   // 8 wave32s
#define BATCHES_PER_WAVE (PTS_PER_TILE / (WAVES_PER_BLOCK * 16))  // 8

typedef float v2fe __attribute__((ext_vector_type(2)));
typedef float v4fe __attribute__((ext_vector_type(4)));
typedef float v8fe __attribute__((ext_vector_type(8)));

struct Consts {
    float wi[9];
    float we[9];
    float be[3];
    float wr[9];
};

// Scalar path (tail handling): out_i = sum_j v_j * W[i][j]  (x @ W.T)
__device__ __forceinline__ float2 transform_point(float x, float y, float z,
                                                  const Consts& c) {
    float q0 = fmaxf(fmaf(c.wi[0], x, fmaf(c.wi[1], y, c.wi[2] * z)), 0.0f);
    float q1 = fmaxf(fmaf(c.wi[3], x, fmaf(c.wi[4], y, c.wi[5] * z)), 0.0f);
    float q2 = fmaxf(fmaf(c.wi[6], x, fmaf(c.wi[7], y, c.wi[8] * z)), 0.0f);

    float r0 = fmaxf(fmaf(c.we[0], q0, fmaf(c.we[1], q1, fmaf(c.we[2], q2, c.be[0]))), 0.0f);
    float r1 = fmaxf(fmaf(c.we[3], q0, fmaf(c.we[4], q1, fmaf(c.we[5], q2, c.be[1]))), 0.0f);
    float r2 = fmaxf(fmaf(c.we[6], q0, fmaf(c.we[7], q1, fmaf(c.we[8], q2, c.be[2]))), 0.0f);

    float s0 = fmaxf(fmaf(c.wr[0], r0, fmaf(c.wr[1], r1, c.wr[2] * r2)), 0.0f);
    float s1 = fmaxf(fmaf(c.wr[3], r0, fmaf(c.wr[4], r1, c.wr[5] * r2)), 0.0f);
    float s2 = fmaxf(fmaf(c.wr[6], r0, fmaf(c.wr[7], r1, c.wr[8] * r2)), 0.0f);

    float inv = 1.0f / s2;
    return make_float2(rintf(s0 * inv), rintf(s1 * inv));
}

// Build this lane's slice of the 16x4 f32 WMMA A-matrix:
//   row M = lane%16 holds [W[M][0], W[M][1], W[M][2], bias[M]]; rows 3..15 = 0.
// A layout (ISA 7.12.2, 32-bit A 16x4): lanes 0-15 -> {K0,K1} in {V0,V1},
// lanes 16-31 -> {K2,K3}.
__device__ __forceinline__ v2fe make_wmma_A(const float* w, float b0, float b1,
                                            float b2, int m, bool lo) {
    float c0 = (m == 0) ? w[0] : (m == 1) ? w[3] : (m == 2) ? w[6] : 0.0f;
    float c1 = (m == 0) ? w[1] : (m == 1) ? w[4] : (m == 2) ? w[7] : 0.0f;
    float c2 = (m == 0) ? w[2] : (m == 1) ? w[5] : (m == 2) ? w[8] : 0.0f;
    float c3 = (m == 0) ? b0   : (m == 1) ? b1   : (m == 2) ? b2   : 0.0f;
    v2fe a;
    a.x = lo ? c0 : c2;
    a.y = lo ? c1 : c3;
    return a;
}

__global__ __launch_bounds__(TPB) void ImageAlignment_proj_kernel(
        const float* __restrict__ pts,    // N x 3
        const float* __restrict__ Winv,   // 3 x 3 row-major
        const float* __restrict__ Wext,   // 3 x 3
        const float* __restrict__ bext,   // 3
        const float* __restrict__ Wrgb,   // 3 x 3
        float* __restrict__ out,          // N x 2
        int n_points, int n_tiles) {
    __shared__ float4 smem[CHUNKS_PER_TILE];

    // Uniform weights -> scalar loads, kept in registers for the whole block.
    Consts c;
#pragma unroll
    for (int i = 0; i < 9; ++i) {
        c.wi[i] = Winv[i];
        c.we[i] = Wext[i];
        c.wr[i] = Wrgb[i];
    }
    c.be[0] = bext[0]; c.be[1] = bext[1]; c.be[2] = bext[2];

    const int t    = threadIdx.x;
    const int lane = t & 31;
    const int wave = t >> 5;
    const int m    = lane & 15;       // matrix row (A) / point column (B)
    const bool lo  = lane < 16;

    // Per-stage A matrices (constant across the whole kernel).
    const v2fe A1 = make_wmma_A(c.wi, 0.0f, 0.0f, 0.0f, m, lo);
    const v2fe A2 = make_wmma_A(c.we, c.be[0], c.be[1], c.be[2], m, lo);
    const v2fe A3 = make_wmma_A(c.wr, 0.0f, 0.0f, 0.0f, m, lo);

    // Generic LDS pointer: low 32 bits are the LDS byte offset.
    const uint32_t lds_base = (uint32_t)(uintptr_t)(&smem[0]);
    const float* sf = (const float*)smem;

    for (int tile = blockIdx.x; tile < n_tiles; tile += gridDim.x) {
        // ---- Stage 12 KB of input into LDS via the CDNA5 async DMA path ----
        const uint64_t gbase =
            (uint64_t)(uintptr_t)pts + (uint64_t)tile * (CHUNKS_PER_TILE * 16u);
#pragma unroll
        for (int k = 0; k < 3; ++k) {
            uint32_t chunk = (uint32_t)t + (uint32_t)k * TPB;   // 0..767
            uint32_t ldsa  = lds_base + chunk * 16u;
            uint32_t voff  = chunk * 16u;                        // byte offset
            asm volatile(
                "global_load_async_to_lds_b128 %0, %1, %2 offset:0"
                :
                : "v"(ldsa), "v"(voff), "s"(gbase)
                : "memory");
        }
        asm volatile("s_wait_asynccnt 0" ::: "memory");  // our wave's DMA done
        __syncthreads();                                 // visible to all waves

        // ---- WMMA pipeline: 16 points per batch, 8 batches per wave ----
#pragma unroll
        for (int j = 0; j < BATCHES_PER_WAVE; ++j) {
            const int bp = wave * (BATCHES_PER_WAVE * 16) + j * 16 + m;

            // B (4x16 f32): V0 = x (lanes 0-15) / z (lanes 16-31),
            //               V1 = y (lanes 0-15) / 1.0 (lanes 16-31)
            float bx = sf[3 * bp + (lo ? 0 : 2)];
            float by = sf[3 * bp + 1];
            v2fe B1;
            B1.x = bx;
            B1.y = lo ? by : 1.0f;

            v8fe cz = {0.f, 0.f, 0.f, 0.f, 0.f, 0.f, 0.f, 0.f};

            // Stage 1: q = relu(Winv @ v)
            v8fe d1 = __builtin_amdgcn_wmma_f32_16x16x4_f32(
                false, A1, false, B1, (short)0, cz, false, false);
            float r0 = fmaxf(d1[0], 0.0f);
            float r1 = fmaxf(d1[1], 0.0f);
            float r2 = fmaxf(d1[2], 0.0f);

            // Repack D -> B: ch0/ch1 already in place; ch2 low->high half.
            float sh2 = __shfl(r2, m, 32);
            v2fe B2;
            B2.x = lo ? r0 : sh2;
            B2.y = lo ? r1 : 1.0f;

            // Stage 2: r = relu(Wext @ q + bext)   (bias via K=3 column of A2)
            v8fe d2 = __builtin_amdgcn_wmma_f32_16x16x4_f32(
                false, A2, false, B2, (short)0, cz, false, false);
            r0 = fmaxf(d2[0], 0.0f);
            r1 = fmaxf(d2[1], 0.0f);
            r2 = fmaxf(d2[2], 0.0f);

            sh2 = __shfl(r2, m, 32);
            v2fe B3;
            B3.x = lo ? r0 : sh2;
            B3.y = lo ? r1 : 1.0f;

            // Stage 3: s = relu(Wrgb @ r)
            v8fe d3 = __builtin_amdgcn_wmma_f32_16x16x4_f32(
                false, A3, false, B3, (short)0, cz, false, false);
            float s0 = fmaxf(d3[0], 0.0f);
            float s1 = fmaxf(d3[1], 0.0f);
            float s2 = fmaxf(d3[2], 0.0f);

            // Perspective divide + round-half-to-even (== jnp.round)
            float inv = 1.0f / s2;
            float ox = rintf(s0 * inv);
            float oy = rintf(s1 * inv);

            // Lanes 0-15 hold the 16 points -> contiguous 128B NT store.
            if (lo) {
                size_t p = (size_t)tile * PTS_PER_TILE + (size_t)bp;
                v2fe o = {ox, oy};
                __builtin_nontemporal_store(o, (v2fe*)out + p);
            }
        }
        __syncthreads();  // LDS reuse fence before next grid-stride tile
    }

    // ---- Tail (n_points not multiple of 1024): direct scalar path ----
    if (blockIdx.x == 0) {
        for (int p = n_tiles * PTS_PER_TILE + t; p < n_points; p += TPB) {
            float x = pts[3 * p + 0];
            float y = pts[3 * p + 1];
            float z = pts[3 * p + 2];
            float2 o = transform_point(x, y, z, c);
            out[2 * p + 0] = o.x;
            out[2 * p + 1] = o.y;
        }
    }
}

extern "C" void kernel_launch(void* const* d_in, const int* in_sizes, int n_in,
                              void* d_out, int out_size, void* d_ws, size_t ws_size,
                              hipStream_t stream) {
    const float* pts  = (const float*)d_in[0];  // (N,3) f32
    const float* Winv = (const float*)d_in[1];  // (3,3) f32
    const float* Wext = (const float*)d_in[2];  // (3,3) f32
    const float* bext = (const float*)d_in[3];  // (3,)  f32
    const float* Wrgb = (const float*)d_in[4];  // (3,3) f32
    float* out = (float*)d_out;                 // (N,2) f32

    const int n_points = in_sizes[0] / 3;
    const int n_tiles  = n_points / PTS_PER_TILE;   // 16384 for N = 16.7M
    int blocks = n_tiles > 0 ? n_tiles : 1;

    ImageAlignment_proj_kernel<<<blocks, TPB, 0, stream>>>(
        pts, Winv, Wext, bext, Wrgb, out, n_points, n_tiles);
}